// SAGE_70403103916590
// MI455X (gfx1250) — compile-verified
//
#include <hip/hip_runtime.h>
#include <hip/hip_bf16.h>
#include <stdint.h>

#define HID 128
#define CLASSES 64

typedef float v2f __attribute__((ext_vector_type(2)));
typedef float v8f __attribute__((ext_vector_type(8)));

// ---------------------------------------------------------------- zero fill
__global__ void sage_zero_kernel(int* __restrict__ p, long long n) {
    long long i = (long long)blockIdx.x * blockDim.x + threadIdx.x;
    if (i < n) p[i] = 0;
}

// ---------------------------------------------------------------- in-degree count
__global__ void sage_count_kernel(const long long* __restrict__ dst,
                                  int* __restrict__ count, int E) {
    int e = blockIdx.x * blockDim.x + threadIdx.x;
    if (e < E) atomicAdd(&count[(int)dst[e]], 1);
}

// ---------------------------------------------------------------- exclusive scan (1 block)
__global__ void __launch_bounds__(1024)
sage_scan_kernel(const int* __restrict__ count, int* __restrict__ rowPtr, int N) {
    __shared__ int buf[1024];
    __shared__ int carry;
    if (threadIdx.x == 0) carry = 0;
    __syncthreads();
    for (int base = 0; base < N; base += 1024) {
        int i = base + (int)threadIdx.x;
        int v = (i < N) ? count[i] : 0;
        buf[threadIdx.x] = v;
        __syncthreads();
        for (int off = 1; off < 1024; off <<= 1) {
            int t = 0;
            if ((int)threadIdx.x >= off) t = buf[threadIdx.x - off];
            __syncthreads();
            if ((int)threadIdx.x >= off) buf[threadIdx.x] += t;
            __syncthreads();
        }
        int c = carry;
        int incl = buf[threadIdx.x];
        if (i < N) rowPtr[i] = c + incl - v;   // exclusive prefix
        __syncthreads();
        if (threadIdx.x == 1023) carry = c + buf[1023];
        __syncthreads();
    }
    if (threadIdx.x == 0) rowPtr[N] = carry;
}

// ---------------------------------------------------------------- CSR fill
__global__ void sage_fill_kernel(const long long* __restrict__ src,
                                 const long long* __restrict__ dst,
                                 const int* __restrict__ rowPtr,
                                 int* __restrict__ cursor,
                                 int* __restrict__ colIdx, int E) {
    int e = blockIdx.x * blockDim.x + threadIdx.x;
    if (e < E) {
        int d = (int)dst[e];
        int pos = atomicAdd(&cursor[d], 1);
        colIdx[rowPtr[d] + pos] = (int)src[e];
    }
}

// ------------------------------------------------- gather-sum: agg[n,:] = sum_{e: dst=n} X[src_e,:]
// One wave per destination node; lane l owns features [4l, 4l+4).
__global__ void __launch_bounds__(256)
sage_gather_kernel(const float* __restrict__ X,
                   const int* __restrict__ rowPtr,
                   const int* __restrict__ colIdx,
                   float* __restrict__ agg, int N) {
    int node = (blockIdx.x * blockDim.x + threadIdx.x) >> 5;
    int lane = threadIdx.x & 31;
    if (node >= N) return;
    int beg = rowPtr[node];
    int end = rowPtr[node + 1];
    float4 a0 = {0.f, 0.f, 0.f, 0.f};
    float4 a1 = {0.f, 0.f, 0.f, 0.f};
    int j = beg;
    for (; j + 1 < end; j += 2) {
        int s0 = colIdx[j];
        int s1 = colIdx[j + 1];
        const float4 v0 = *(const float4*)(X + (long long)s0 * HID + lane * 4);
        const float4 v1 = *(const float4*)(X + (long long)s1 * HID + lane * 4);
        a0.x += v0.x; a0.y += v0.y; a0.z += v0.z; a0.w += v0.w;
        a1.x += v1.x; a1.y += v1.y; a1.z += v1.z; a1.w += v1.w;
    }
    if (j < end) {
        int s0 = colIdx[j];
        const float4 v0 = *(const float4*)(X + (long long)s0 * HID + lane * 4);
        a0.x += v0.x; a0.y += v0.y; a0.z += v0.z; a0.w += v0.w;
    }
    a0.x += a1.x; a0.y += a1.y; a0.z += a1.z; a0.w += a1.w;
    *(float4*)(agg + (long long)node * HID + lane * 4) = a0;
}

// ---------------------------------------------------------------- fused layer
// H[i,:] = relu( X[i,:] @ W + b + agg[i,:] / max(deg_i,1) ),  deg from rowPtr.
// Block = 256 threads = 8 waves; wave w -> output cols [16w,16w+16) via
// V_WMMA_F32_16X16X4_F32, K-loop over 128. 50000 % 16 == 0 -> EXEC all-1s.
__global__ void __launch_bounds__(256)
sage_layer_kernel(const float* __restrict__ X,      // [N,128]
                  const float* __restrict__ W,      // [128,128] row-major (in,out)
                  const float* __restrict__ b,      // [128]
                  const float* __restrict__ agg,    // [N,128]
                  const int*   __restrict__ rowPtr, // [N+1]
                  float* __restrict__ H)            // [N,128]
{
    const int wave   = threadIdx.x >> 5;
    const int lane   = threadIdx.x & 31;
    const int half   = lane >> 4;
    const int lane16 = lane & 15;
    const int mBase  = blockIdx.x * 16;
    const int nBase  = wave * 16;

    v8f acc = {};
    const float* arow = X + (long long)(mBase + lane16) * HID;

    for (int k = 0; k < HID; k += 4) {
        // A fragment (16x4 f32): lanes 0-15 hold (k,k+1); lanes 16-31 hold (k+2,k+3)
        v2f a = *(const v2f*)(arow + k + half * 2);
        // B fragment (4x16 f32): VGPR0 = W[k+2*half][col], VGPR1 = W[k+1+2*half][col]
        const float* bp = W + (long long)(k + half * 2) * HID + nBase + lane16;
        v2f bf;
        bf.x = bp[0];
        bf.y = bp[HID];
        acc = __builtin_amdgcn_wmma_f32_16x16x4_f32(
            false, a, false, bf, (short)0, acc, false, false);
    }

    const int col = nBase + lane16;
    const float bias = b[col];
#pragma unroll
    for (int v = 0; v < 8; ++v) {
        int row = mBase + v + half * 8;
        int cnt = rowPtr[row + 1] - rowPtr[row];
        float inv = 1.0f / (float)((cnt > 1) ? cnt : 1);
        float val = acc[v] + bias + agg[(long long)row * HID + col] * inv;
        H[(long long)row * HID + col] = fmaxf(val, 0.0f);
    }
}

// ---------------------------------------------------------------- mean/max readout
__global__ void __launch_bounds__(256)
sage_reduce_kernel(const float* __restrict__ H, float* __restrict__ gsum,
                   unsigned int* __restrict__ gmax, int N) {
    int col  = threadIdx.x & (HID - 1);
    int rsub = threadIdx.x >> 7;
    float s = 0.0f, m = 0.0f;                 // post-relu => >= 0
    for (int row = blockIdx.x * 2 + rsub; row < N; row += gridDim.x * 2) {
        float v = H[(long long)row * HID + col];
        s += v;
        m = fmaxf(m, v);
    }
    __shared__ float ls[256];
    __shared__ float lm[256];
    ls[threadIdx.x] = s;
    lm[threadIdx.x] = m;
    __syncthreads();
    if (rsub == 0) {
        s += ls[threadIdx.x + 128];
        m = fmaxf(m, lm[threadIdx.x + 128]);
        atomicAdd(&gsum[col], s);
        atomicMax(&gmax[col], __float_as_uint(m));  // valid for non-negative floats
    }
}

// ---------------------------------------------------------------- head GEMV
__global__ void __launch_bounds__(64)
sage_head_kernel(const float* __restrict__ gsum, const unsigned int* __restrict__ gmax,
                 const float* __restrict__ Wout, const float* __restrict__ bout,
                 float* __restrict__ out, float invN) {
    int c = threadIdx.x;
    float acc = bout[c];
#pragma unroll 4
    for (int k = 0; k < HID; ++k)
        acc += (gsum[k] * invN) * Wout[k * CLASSES + c];
#pragma unroll 4
    for (int k = 0; k < HID; ++k)
        acc += __uint_as_float(gmax[k]) * Wout[(HID + k) * CLASSES + c];
    out[c] = acc;
}

// ---------------------------------------------------------------- launcher
extern "C" void kernel_launch(void* const* d_in, const int* in_sizes, int n_in,
                              void* d_out, int out_size, void* d_ws, size_t ws_size,
                              hipStream_t stream) {
    const float*     x    = (const float*)d_in[0];
    const long long* ei   = (const long long*)d_in[1];   // int64 per reference
    const float*     W1   = (const float*)d_in[2];
    const float*     b1   = (const float*)d_in[3];
    const float*     W2   = (const float*)d_in[4];
    const float*     b2   = (const float*)d_in[5];
    const float*     Wout = (const float*)d_in[6];
    const float*     bout = (const float*)d_in[7];
    float*           out  = (float*)d_out;

    const int N = in_sizes[0] / HID;     // 50000
    const int E = in_sizes[1] / 2;       // 800000
    const long long NH = (long long)N * HID;

    const long long* src = ei;
    const long long* dst = ei + E;

    // workspace layout: count | cursor | gsum | gmax | rowPtr | colIdx | (align16) agg | h1 | h2
    int*          count  = (int*)d_ws;                 // N
    int*          cursor = count + N;                  // N
    float*        gsum   = (float*)(cursor + N);       // 128
    unsigned int* gmax   = (unsigned int*)(gsum + HID);// 128
    int*          rowPtr = (int*)(gmax + HID);         // N+1
    int*          colIdx = rowPtr + N + 1;             // E
    float*        agg    = (float*)(((uintptr_t)(colIdx + E) + 15) & ~(uintptr_t)15);
    float*        h1     = agg + NH;
    float*        h2     = h1 + NH;

    // 1) zero count, cursor, gsum, gmax (contiguous ints/words)
    {
        long long n = 2LL * N + 2 * HID;
        sage_zero_kernel<<<(int)((n + 255) / 256), 256, 0, stream>>>(count, n);
    }
    // 2) in-degree counts
    sage_count_kernel<<<(E + 255) / 256, 256, 0, stream>>>(dst, count, E);
    // 3) exclusive scan -> rowPtr
    sage_scan_kernel<<<1, 1024, 0, stream>>>(count, rowPtr, N);
    // 4) CSR column fill
    sage_fill_kernel<<<(E + 255) / 256, 256, 0, stream>>>(src, dst, rowPtr, cursor, colIdx, E);
    // 5) agg(x) by gather
    sage_gather_kernel<<<(N * 32 + 255) / 256, 256, 0, stream>>>(x, rowPtr, colIdx, agg, N);
    // 6) h1 = relu(x@W1 + b1 + agg/deg)
    sage_layer_kernel<<<N / 16, 256, 0, stream>>>(x, W1, b1, agg, rowPtr, h1);
    // 7) agg(h1) by gather (full overwrite, no re-zero needed)
    sage_gather_kernel<<<(N * 32 + 255) / 256, 256, 0, stream>>>(h1, rowPtr, colIdx, agg, N);
    // 8) h2 = relu(h1@W2 + b2 + agg/deg)
    sage_layer_kernel<<<N / 16, 256, 0, stream>>>(h1, W2, b2, agg, rowPtr, h2);
    // 9) column-wise mean/max
    sage_reduce_kernel<<<512, 256, 0, stream>>>(h2, gsum, gmax, N);
    // 10) head
    sage_head_kernel<<<1, 64, 0, stream>>>(gsum, gmax, Wout, bout, out, 1.0f / (float)N);
}